// HeteroGNN_83099027243503
// MI455X (gfx1250) — compile-verified
//
#include <hip/hip_runtime.h>

#define HDIM 128
#define NP_N 200000
#define NA_N 100000
#define NV_N 10000
#define E_WR 2000000
#define E_CI 2000000
#define E_PU 200000

typedef __attribute__((ext_vector_type(2))) float v2f;
typedef __attribute__((ext_vector_type(8))) float v8f;

// ---------------------------------------------------------------- zero fill
__global__ void zero2_kernel(float* p0, long long n0, float* p1, long long n1) {
  long long i = (long long)blockIdx.x * blockDim.x + threadIdx.x;
  long long stride = (long long)gridDim.x * blockDim.x;
  for (long long j = i; j < n0; j += stride) p0[j] = 0.0f;
  for (long long j = i; j < n1; j += stride) p1[j] = 0.0f;
}

// ------------------------------------------------- edge scatter (segment sum)
// One wave32 per edge: each lane gathers a float4 of the 128-float source row
// and atomically accumulates it into the destination row. Lane 0 counts.
__global__ void scatter_mean_kernel(const float* __restrict__ xsrc,
                                    const int* __restrict__ ei, int E,
                                    float* __restrict__ agg,
                                    float* __restrict__ cnt) {
  long long t = (long long)blockIdx.x * blockDim.x + threadIdx.x;
  int e = (int)(t >> 5);
  int lane = (int)(t & 31);
  if (e >= E) return;
  int s = ei[e];                 // src row  (ei[0, :])
  int d = ei[(long long)E + e];  // dst row  (ei[1, :])
  const float4* sp = (const float4*)(xsrc + (size_t)s * HDIM);
  float4 v = sp[lane];
  float* ap = agg + (size_t)d * HDIM + lane * 4;
  __hip_atomic_fetch_add(ap + 0, v.x, __ATOMIC_RELAXED, __HIP_MEMORY_SCOPE_AGENT);
  __hip_atomic_fetch_add(ap + 1, v.y, __ATOMIC_RELAXED, __HIP_MEMORY_SCOPE_AGENT);
  __hip_atomic_fetch_add(ap + 2, v.z, __ATOMIC_RELAXED, __HIP_MEMORY_SCOPE_AGENT);
  __hip_atomic_fetch_add(ap + 3, v.w, __ATOMIC_RELAXED, __HIP_MEMORY_SCOPE_AGENT);
  if (lane == 0)
    __hip_atomic_fetch_add(cnt + d, 1.0f, __ATOMIC_RELAXED, __HIP_MEMORY_SCOPE_AGENT);
}

// --------------------------------------------------- fused WMMA f32 GEMM
// out[rows x 128] (+)= rowscale(A) @ W + bias
// flags bit0: scale each A row by 1/max(cnt[row],1)   (segment mean)
// flags bit1: accumulate into existing out
// flags bit2: add bias (broadcast over rows)
// One wave computes a 16x16 tile via 32x V_WMMA_F32_16X16X4_F32 (K=128).
// 4 waves/block cover 64 cols; gridDim.y=2 covers all 128 cols.
__global__ __launch_bounds__(128) void gemm128_wmma_kernel(
    const float* __restrict__ A, const float* __restrict__ cnt,
    const float* __restrict__ W, const float* __restrict__ bias,
    float* __restrict__ out, int rows, int flags) {
  const int lane = threadIdx.x & 31;
  const int wave = threadIdx.x >> 5;
  const int half = lane >> 4;   // 0: lanes 0-15, 1: lanes 16-31
  const int lm   = lane & 15;
  const int row0 = blockIdx.x * 16;
  const int col0 = (blockIdx.y * 4 + wave) * 16;

  // Pull the 64KB weight matrix toward the caches (global_prefetch_b8).
  __builtin_prefetch(W + (size_t)threadIdx.x * 128, 0, 1);

  int r = row0 + lm;
  if (r >= rows) r = rows - 1;  // clamp; masked on store
  const int c = col0 + lm;

  float rscale = 1.0f;
  if (flags & 1) rscale = 1.0f / fmaxf(cnt[r], 1.0f);

  const float2* __restrict__ Arow = (const float2*)(A + (size_t)r * HDIM);

  // C/D layout: VGPR v holds M = v + 8*half, N = lm.
  v8f acc;
#pragma unroll
  for (int v = 0; v < 8; ++v) {
    int rr = row0 + v + 8 * half;
    float cv = 0.0f;
    if ((flags & 2) && rr < rows) cv = out[(size_t)rr * HDIM + c];
    if (flags & 4) cv += bias[c];
    acc[v] = cv;
  }

  // A 16x4 f32 layout: lane holds A[M=lm][K = k0 + 2*half + {0,1}] (float2).
  // B 4x16 f32 layout: lane holds W[K = k0 + 2*half + {0,1}][N = lm].
#pragma unroll
  for (int k0 = 0; k0 < HDIM; k0 += 4) {
    float2 a2 = Arow[(k0 >> 1) + half];
    v2f a;
    a.x = a2.x * rscale;
    a.y = a2.y * rscale;
    const int kb = k0 + 2 * half;
    v2f b;
    b.x = W[(size_t)kb * HDIM + c];
    b.y = W[(size_t)(kb + 1) * HDIM + c];
    acc = __builtin_amdgcn_wmma_f32_16x16x4_f32(
        /*neg_a=*/false, a, /*neg_b=*/false, b,
        /*c_mod=*/(short)0, acc, /*reuse_a=*/false, /*reuse_b=*/false);
  }

#pragma unroll
  for (int v = 0; v < 8; ++v) {
    int rr = row0 + v + 8 * half;
    if (rr < rows) out[(size_t)rr * HDIM + c] = acc[v];
  }
}

// ---------------------------------------------------------------- helpers
__global__ void wsum3_kernel(const float* __restrict__ a, const float* __restrict__ b,
                             const float* __restrict__ c, float* __restrict__ o, int n) {
  int i = blockIdx.x * blockDim.x + threadIdx.x;
  if (i < n) o[i] = a[i] + b[i] + c[i];
}

__global__ void relu_scale_kernel(float* p, long long n, float s) {
  long long i = (long long)blockIdx.x * blockDim.x + threadIdx.x;
  long long stride = (long long)gridDim.x * blockDim.x;
  for (long long j = i; j < n; j += stride) p[j] = fmaxf(p[j] * s, 0.0f);
}

// ---------------------------------------------------------------- driver
extern "C" void kernel_launch(void* const* d_in, const int* in_sizes, int n_in,
                              void* d_out, int out_size, void* d_ws, size_t ws_size,
                              hipStream_t stream) {
  const float* xp0 = (const float*)d_in[0];
  const float* xa0 = (const float*)d_in[1];
  const float* xv0 = (const float*)d_in[2];
  const float* Wl  = (const float*)d_in[3];  // [2,5,128,128]
  const float* bl  = (const float*)d_in[4];  // [2,5,128]
  const float* Wr  = (const float*)d_in[5];  // [2,5,128,128]
  const int* eiw  = (const int*)d_in[6];   // [2, E]
  const int* eic  = (const int*)d_in[7];
  const int* eip  = (const int*)d_in[8];
  const int* eirw = (const int*)d_in[9];
  const int* eirp = (const int*)d_in[10];

  float* outp   = (float*)d_out;
  float* out_xp = outp;
  float* out_xa = outp + (size_t)NP_N * HDIM;
  float* out_xv = out_xa + (size_t)NA_N * HDIM;

  // workspace carve
  float* ws   = (float*)d_ws;
  float* agg  = ws;                              // NP*H (max over node types)
  float* cnt  = agg + (size_t)NP_N * HDIM;       // NP
  float* xp1  = cnt + NP_N;                      // NP*H (layer-1 paper feats)
  float* xa1  = xp1 + (size_t)NP_N * HDIM;       // NA*H
  float* xv1  = xa1 + (size_t)NA_N * HDIM;       // NV*H
  float* wsum = xv1 + (size_t)NV_N * HDIM;       // H*H

  const size_t HH = (size_t)HDIM * HDIM;

  auto zero2 = [&](float* a, long long na, float* b, long long nb) {
    zero2_kernel<<<2048, 256, 0, stream>>>(a, na, b, nb);
  };
  auto scat = [&](const float* xs, const int* ei, int E) {
    long long threads = (long long)E * 32;
    int blocks = (int)((threads + 255) / 256);
    scatter_mean_kernel<<<blocks, 256, 0, stream>>>(xs, ei, E, agg, cnt);
  };
  auto gemm = [&](float* o, const float* A, const float* cn, const float* Wm,
                  const float* b, int rows, int flags) {
    dim3 g((rows + 15) / 16, 2);
    gemm128_wmma_kernel<<<g, 128, 0, stream>>>(A, cn, Wm, b, o, rows, flags);
  };

  for (int l = 0; l < 2; ++l) {
    const float* sp = l ? xp1 : xp0;
    const float* sa = l ? xa1 : xa0;
    const float* sv = l ? xv1 : xv0;
    float* dp = l ? out_xp : xp1;
    float* da = l ? out_xa : xa1;
    float* dv = l ? out_xv : xv1;

    // ---- authors: relation 3 (rev_writes: paper -> author) ----
    zero2(agg, (long long)NA_N * HDIM, cnt, NA_N);
    scat(sp, eirw, E_WR);
    gemm(da, agg, cnt, Wl + (l * 5 + 3) * HH, bl + (l * 5 + 3) * HDIM, NA_N, 1 | 4);
    gemm(da, sa, nullptr, Wr + (l * 5 + 3) * HH, nullptr, NA_N, 2);
    relu_scale_kernel<<<2048, 256, 0, stream>>>(da, (long long)NA_N * HDIM, 1.0f);

    // ---- venues: relation 4 (rev_publishes: paper -> venue) ----
    zero2(agg, (long long)NV_N * HDIM, cnt, NV_N);
    scat(sp, eirp, E_PU);
    gemm(dv, agg, cnt, Wl + (l * 5 + 4) * HH, bl + (l * 5 + 4) * HDIM, NV_N, 1 | 4);
    gemm(dv, sv, nullptr, Wr + (l * 5 + 4) * HH, nullptr, NV_N, 2);
    relu_scale_kernel<<<2048, 256, 0, stream>>>(dv, (long long)NV_N * HDIM, 1.0f);

    // ---- papers: relations 0 (writes), 1 (cites), 2 (publishes) ----
    zero2(agg, (long long)NP_N * HDIM, cnt, NP_N);
    scat(sa, eiw, E_WR);
    gemm(dp, agg, cnt, Wl + (l * 5 + 0) * HH, bl + (l * 5 + 0) * HDIM, NP_N, 1 | 4);

    zero2(agg, (long long)NP_N * HDIM, cnt, NP_N);
    scat(sp, eic, E_CI);
    gemm(dp, agg, cnt, Wl + (l * 5 + 1) * HH, bl + (l * 5 + 1) * HDIM, NP_N, 1 | 2 | 4);

    zero2(agg, (long long)NP_N * HDIM, cnt, NP_N);
    scat(sv, eip, E_PU);
    gemm(dp, agg, cnt, Wl + (l * 5 + 2) * HH, bl + (l * 5 + 2) * HDIM, NP_N, 1 | 2 | 4);

    // self-loop terms share x_dst = papers: xp @ (Wr0 + Wr1 + Wr2)
    wsum3_kernel<<<(int)((HH + 255) / 256), 256, 0, stream>>>(
        Wr + (l * 5 + 0) * HH, Wr + (l * 5 + 1) * HH, Wr + (l * 5 + 2) * HH,
        wsum, (int)HH);
    gemm(dp, sp, nullptr, wsum, nullptr, NP_N, 2);
    relu_scale_kernel<<<2048, 256, 0, stream>>>(dp, (long long)NP_N * HDIM, 1.0f / 3.0f);
  }
}